// BinsCombinerLayer_36747740185227
// MI455X (gfx1250) — compile-verified
//
#include <hip/hip_runtime.h>

// out[b] = (1/S) * sum_s cumsum_b(in*cen)[s,b]  ==  cumsum_b( colsum )[b] / S
// S = 8192 sets, B = 4096 bins. Memory-bound: 256 MB in -> ~11 us at 23.3 TB/s.

#define NUM_SETS 8192
#define NUM_BINS 4096

typedef __attribute__((ext_vector_type(2))) float v2f;
typedef __attribute__((ext_vector_type(8))) float v8f;

// ---------------------------------------------------------------------------
// Phase 1: streaming column partial sums (bandwidth-limited, float4 coalesced).
// Grid: (4, chunks) x 256 threads. Block (x,y) covers columns
// [x*1024, x*1024+1024) and rows [y*rpc, (y+1)*rpc).
// ---------------------------------------------------------------------------
__global__ void bins_colsum_partials(const float* __restrict__ in,
                                     const float* __restrict__ cen,
                                     float* __restrict__ part,   // [chunks][4096]
                                     int rows_per_chunk) {
  const int col4 = blockIdx.x * 256 + threadIdx.x;   // float4 slot 0..1023
  const int r0   = blockIdx.y * rows_per_chunk;
  const int r1   = r0 + rows_per_chunk;

  const float4* __restrict__ in4  = (const float4*)in;
  const float4* __restrict__ cen4 = (const float4*)cen;

  float4 acc = make_float4(0.f, 0.f, 0.f, 0.f);
  size_t idx = (size_t)r0 * (NUM_BINS / 4) + col4;

  for (int r = r0; r < r1; ++r, idx += NUM_BINS / 4) {
    if (r + 16 < r1) {
      // CDNA5 speculative prefetch (global_prefetch_b8), 16 rows ahead
      __builtin_prefetch(&in4[idx + 16 * (NUM_BINS / 4)], 0, 1);
      __builtin_prefetch(&cen4[idx + 16 * (NUM_BINS / 4)], 0, 1);
    }
    float4 a = in4[idx];
    float4 c = cen4[idx];
    acc.x = fmaf(a.x, c.x, acc.x);
    acc.y = fmaf(a.y, c.y, acc.y);
    acc.z = fmaf(a.z, c.z, acc.z);
    acc.w = fmaf(a.w, c.w, acc.w);
  }
  ((float4*)part)[(size_t)blockIdx.y * (NUM_BINS / 4) + col4] = acc;
}

// ---------------------------------------------------------------------------
// Phase 2: WMMA reduction of the `chunks` partial rows down to colsum[4096].
// A = ones(16x4)  =>  D[m,n] = sum_k B[k,n] + C[m,n]: a 4-row column reduction
// per v_wmma_f32_16x16x4_f32, invariant to the K-row <-> VGPR assignment of B.
// One wave per 16-column tile; 32 chained accumulating WMMAs for chunks=128.
// D layout (documented): VGPR0, lanes 0..15 = row M=0, N=lane -> read c[0].
// ---------------------------------------------------------------------------
__global__ void bins_wmma_reduce(const float* __restrict__ part,  // [chunks][4096]
                                 float* __restrict__ colsum,      // [4096]
                                 int chunks) {                     // multiple of 4
  const int lane = threadIdx.x;      // 0..31, full wave, EXEC all ones
  const int l16  = lane >> 4;        // which half-wave
  const int lm   = lane & 15;        // N within tile
  const int c0   = blockIdx.x * 16;  // tile base column

  v2f a; a.x = 1.0f; a.y = 1.0f;     // ones A-matrix (layout-agnostic)
  v8f c = {};

  for (int r = 0; r < chunks; r += 4) {
    v2f b;
    b.x = part[(size_t)(r + l16)     * NUM_BINS + c0 + lm];
    b.y = part[(size_t)(r + 2 + l16) * NUM_BINS + c0 + lm];
    // (neg_a, A, neg_b, B, c_mod, C, reuse_a, reuse_b)
    c = __builtin_amdgcn_wmma_f32_16x16x4_f32(false, a, false, b,
                                              (short)0, c, false, false);
  }
  if (lane < 16) colsum[c0 + lm] = c[0];
}

// ---------------------------------------------------------------------------
// Phase 3: single-workgroup inclusive scan of colsum[4096], scale by 1/S.
// 256 threads x 16 contiguous elements each; LDS Hillis-Steele over totals.
// ---------------------------------------------------------------------------
__global__ void bins_scan_finalize(const float* __restrict__ colsum,
                                   float* __restrict__ out) {
  __shared__ float s[256];
  const int t = threadIdx.x;

  float v[16];
  const float4* cs4 = (const float4*)(colsum + t * 16);
#pragma unroll
  for (int i = 0; i < 4; ++i) {
    float4 x = cs4[i];
    v[4 * i + 0] = x.x; v[4 * i + 1] = x.y;
    v[4 * i + 2] = x.z; v[4 * i + 3] = x.w;
  }
  // serial inclusive cumsum of this thread's 16 elements
  float run = 0.f;
#pragma unroll
  for (int i = 0; i < 16; ++i) { run += v[i]; v[i] = run; }

  s[t] = run;
  __syncthreads();
#pragma unroll
  for (int off = 1; off < 256; off <<= 1) {
    float add = (t >= off) ? s[t - off] : 0.f;
    __syncthreads();
    s[t] += add;
    __syncthreads();
  }
  const float excl  = s[t] - run;          // exclusive prefix of thread totals
  const float scale = 1.0f / (float)NUM_SETS;

  float4* out4 = (float4*)(out + t * 16);
#pragma unroll
  for (int i = 0; i < 4; ++i) {
    float4 x;
    x.x = (v[4 * i + 0] + excl) * scale;
    x.y = (v[4 * i + 1] + excl) * scale;
    x.z = (v[4 * i + 2] + excl) * scale;
    x.w = (v[4 * i + 3] + excl) * scale;
    out4[i] = x;
  }
}

extern "C" void kernel_launch(void* const* d_in, const int* in_sizes, int n_in,
                              void* d_out, int out_size, void* d_ws, size_t ws_size,
                              hipStream_t stream) {
  (void)in_sizes; (void)n_in; (void)out_size;

  const float* in  = (const float*)d_in[0];
  const float* cen = (const float*)d_in[1];
  float* out = (float*)d_out;

  // Workspace layout: part[chunks][4096] followed by colsum[4096].
  int chunks = 128;  // power of two, multiple of 4
  while ((size_t)chunks * NUM_BINS * sizeof(float) + NUM_BINS * sizeof(float) > ws_size
         && chunks > 4) {
    chunks >>= 1;
  }
  const int rows_per_chunk = NUM_SETS / chunks;

  float* part   = (float*)d_ws;
  float* colsum = part + (size_t)chunks * NUM_BINS;

  dim3 g1(NUM_BINS / (256 * 4), chunks);           // (4, chunks)
  bins_colsum_partials<<<g1, 256, 0, stream>>>(in, cen, part, rows_per_chunk);

  bins_wmma_reduce<<<NUM_BINS / 16, 32, 0, stream>>>(part, colsum, chunks);

  bins_scan_finalize<<<1, 256, 0, stream>>>(colsum, out);
}